// APPNP_78426102825064
// MI455X (gfx1250) — compile-verified
//
#include <hip/hip_runtime.h>

#define N_NODES 100000
#define N_EDGES 3200000
#define IN_F    500
#define HID     64
#define OUT_F   64
#define ALPHA   0.1f

typedef __attribute__((ext_vector_type(2))) float v2f;
typedef __attribute__((ext_vector_type(8))) float v8f;

// ---------------------------------------------------------------------------
// Tiled GEMM on the CDNA5 matrix pipe: C[16-row tile, 64 cols] per wave.
// D = A(16xK) * B(Kx64) + bias, optional ReLU.  fp32 WMMA (exact semantics).
// A-frag layout (ISA 7.12.2): lanes 0-15 -> K={k,k+1}, lanes 16-31 -> K={k+2,k+3},
// M = lane&15.  B/C/D: rows striped across lanes within a VGPR.
// ---------------------------------------------------------------------------
template <int KDIM, bool RELU>
__global__ void __launch_bounds__(256)
gemm_wmma_kernel(const float* __restrict__ A, const float* __restrict__ B,
                 const float* __restrict__ bias, float* __restrict__ C) {
    const int lane    = threadIdx.x & 31;
    const int wave    = threadIdx.x >> 5;
    const int rowTile = blockIdx.x * 8 + wave;        // 16 output rows per wave
    if (rowTile >= N_NODES / 16) return;              // wave-uniform: EXEC stays all-1s

    const int nlo   = lane & 15;                      // M for A-frag, N for B-frag
    const int khalf = lane >> 4;                      // 0 -> K pair {0,1}, 1 -> {2,3}

    const float* Arow = A + (size_t)(rowTile * 16 + nlo) * KDIM;

    v8f acc0 = {}, acc1 = {}, acc2 = {}, acc3 = {};

    for (int k = 0; k < KDIM; k += 4) {
        const int ka = k + 2 * khalf;
        v2f a;
        a.x = Arow[ka + 0];                           // contiguous -> b64 load
        a.y = Arow[ka + 1];

        const float* Bp = B + (size_t)ka * HID + nlo; // B[ka][n], B[ka+1][n]
        v2f b0, b1v, b2v, b3v;
        b0.x  = Bp[0];        b0.y  = Bp[HID];
        b1v.x = Bp[16];       b1v.y = Bp[HID + 16];
        b2v.x = Bp[32];       b2v.y = Bp[HID + 32];
        b3v.x = Bp[48];       b3v.y = Bp[HID + 48];

        acc0 = __builtin_amdgcn_wmma_f32_16x16x4_f32(false, a, false, b0,  (short)0, acc0, false, false);
        acc1 = __builtin_amdgcn_wmma_f32_16x16x4_f32(false, a, false, b1v, (short)0, acc1, false, false);
        acc2 = __builtin_amdgcn_wmma_f32_16x16x4_f32(false, a, false, b2v, (short)0, acc2, false, false);
        acc3 = __builtin_amdgcn_wmma_f32_16x16x4_f32(false, a, false, b3v, (short)0, acc3, false, false);
    }

    // C/D layout: VGPR v, lanes 0-15 -> M=v, lanes 16-31 -> M=v+8; N = nlo per 16-col tile
    const float bs0 = bias[nlo];
    const float bs1 = bias[nlo + 16];
    const float bs2 = bias[nlo + 32];
    const float bs3 = bias[nlo + 48];
    const int mbase = rowTile * 16 + 8 * khalf;

#pragma unroll
    for (int v = 0; v < 8; ++v) {
        float* Crow = C + (size_t)(mbase + v) * OUT_F + nlo;
        float x0 = acc0[v] + bs0;
        float x1 = acc1[v] + bs1;
        float x2 = acc2[v] + bs2;
        float x3 = acc3[v] + bs3;
        if (RELU) {
            x0 = fmaxf(x0, 0.0f); x1 = fmaxf(x1, 0.0f);
            x2 = fmaxf(x2, 0.0f); x3 = fmaxf(x3, 0.0f);
        }
        Crow[0]  = x0;
        Crow[16] = x1;
        Crow[32] = x2;
        Crow[48] = x3;
    }
}

// ---------------------------------------------------------------------------
// Seed each propagation step: dst = ALPHA * h   (streaming, float4)
// ---------------------------------------------------------------------------
__global__ void __launch_bounds__(256)
seed_kernel(const float4* __restrict__ h, float4* __restrict__ dst, int n4) {
    int i = blockIdx.x * blockDim.x + threadIdx.x;
    if (i < n4) {
        float4 v = h[i];
        v.x *= ALPHA; v.y *= ALPHA; v.z *= ALPHA; v.w *= ALPHA;
        dst[i] = v;
    }
}

// ---------------------------------------------------------------------------
// Edge scatter: one wave32 per edge, 2 channels per lane.
// acc[dst] += (1-ALPHA) * w * z[src]   (global_atomic_add_f32; hot data in L2)
// ---------------------------------------------------------------------------
__global__ void __launch_bounds__(256)
scatter_kernel(const int* __restrict__ esrc, const int* __restrict__ edst,
               const float* __restrict__ ew, const float* __restrict__ z,
               float* __restrict__ acc) {
    const int lane = threadIdx.x & 31;
    const long long e = (long long)blockIdx.x * 8 + (threadIdx.x >> 5);
    if (e >= N_EDGES) return;

    const int   s = esrc[e];
    const int   d = edst[e];
    const float w = ew[e] * (1.0f - ALPHA);

    const float2 v = ((const float2*)(z + (size_t)s * OUT_F))[lane];
    float* ap = acc + (size_t)d * OUT_F + lane * 2;
    atomicAdd(ap + 0, w * v.x);
    atomicAdd(ap + 1, w * v.y);
}

// ---------------------------------------------------------------------------
extern "C" void kernel_launch(void* const* d_in, const int* in_sizes, int n_in,
                              void* d_out, int out_size, void* d_ws, size_t ws_size,
                              hipStream_t stream) {
    const float* x    = (const float*)d_in[0];
    const int*   esrc = (const int*)  d_in[1];
    const int*   edst = (const int*)  d_in[2];
    const float* ew   = (const float*)d_in[3];
    const float* W1   = (const float*)d_in[4];
    const float* b1   = (const float*)d_in[5];
    const float* W2   = (const float*)d_in[6];
    const float* b2   = (const float*)d_in[7];
    float* out = (float*)d_out;

    const size_t NF = (size_t)N_NODES * OUT_F;   // elements per node-feature buffer
    float* h = (float*)d_ws;                     // encoder output
    float* P = h + NF;                           // ping (also holds y1 transiently)
    float* Q = P + NF;                           // pong

    dim3 blk(256);
    const int gemmBlocks = (N_NODES / 16 + 7) / 8;

    // h = relu(x@W1 + b1) @ W2 + b2  (y1 staged in P, then overwritten)
    gemm_wmma_kernel<IN_F, true ><<<gemmBlocks, blk, 0, stream>>>(x, W1, b1, P);
    gemm_wmma_kernel<HID,  false><<<gemmBlocks, blk, 0, stream>>>(P, W2, b2, h);

    const int n4         = (int)(NF / 4);
    const int seedBlocks = (n4 + 255) / 256;
    const int edgeBlocks = (N_EDGES + 7) / 8;    // 8 edges (waves) per block

    // APPNP: z_{k+1} = (1-a) * A z_k + a * h ; z_0 = h ; last step lands in d_out
    const float* cur = h;
    for (int it = 0; it < 10; ++it) {
        float* nxt = (it == 9) ? out : ((it & 1) ? Q : P);
        seed_kernel<<<seedBlocks, blk, 0, stream>>>((const float4*)h, (float4*)nxt, n4);
        scatter_kernel<<<edgeBlocks, blk, 0, stream>>>(esrc, edst, ew, cur, nxt);
        cur = nxt;
    }
}